// yoloLoss_16930761081398
// MI455X (gfx1250) — compile-verified
//
#include <hip/hip_runtime.h>
#include <hip/hip_bf16.h>

typedef __attribute__((ext_vector_type(2))) float v2f;
typedef __attribute__((ext_vector_type(8))) float v8f;

#define SGRID 14
#define SS (SGRID * SGRID)
#define MAXOBJ 20
#define NCLS 20

// ---------------------------------------------------------------------------
// Wave32 8-accumulator cross-lane reduction using V_WMMA_F32_16X16X4_F32.
// Precondition: called with all 32 lanes of the wave active (EXEC == ~0).
// Step 1: xor-folds 1,2,4 -> each lane holds the sum over its aligned group
//         of 8 lanes, for all 8 accumulators.
// Step 2: lane L contributes A.x = groupsum[L & 7], A.y = 0. Per the ISA
//         A-matrix 16x4 f32 layout (lanes 0-15 = K0/K1, lanes 16-31 = K2/K3):
//           row m (0..7):  A[m,0]=G0[m], A[m,2]=G2[m]
//           row m (8..15): A[m,0]=G1[m-8], A[m,2]=G3[m-8]
//         With B = all-ones (layout-independent), D[m,n] = sum_k A[m,k], and
//         the documented D layout gives lanes<16: d[j]=G0[j]+G2[j],
//         lanes>=16: d[j]=G1[j]+G3[j].
// Step 3: one xor-fold by 16 completes the totals on every lane.
// ---------------------------------------------------------------------------
__device__ __forceinline__ void wave_reduce8_wmma(float acc[8]) {
#pragma unroll
  for (int m = 1; m <= 4; m <<= 1) {
#pragma unroll
    for (int a = 0; a < 8; ++a) acc[a] += __shfl_xor(acc[a], m, 32);
  }
  const int lane = (int)(threadIdx.x & 31u);
  v2f A;
  A.x = acc[lane & 7];
  A.y = 0.0f;
  v2f B;
  B.x = 1.0f;
  B.y = 1.0f;
  v8f C = {0.f, 0.f, 0.f, 0.f, 0.f, 0.f, 0.f, 0.f};
  v8f D = __builtin_amdgcn_wmma_f32_16x16x4_f32(
      /*neg_a=*/false, A, /*neg_b=*/false, B,
      /*c_mod=*/(short)0, C, /*reuse_a=*/false, /*reuse_b=*/false);
#pragma unroll
  for (int a = 0; a < 8; ++a) {
    float t = D[a];
    acc[a] = t + __shfl_xor(t, 16, 32);
  }
}

// ---------------------------------------------------------------------------
// K0: zero conf (10*NCELL f32) + class bitmask (NCELL u32) = 11*NCELL words.
// ---------------------------------------------------------------------------
__global__ void k_zero(unsigned* __restrict__ ws, int words) {
  int i = blockIdx.x * blockDim.x + threadIdx.x;
  if (i < words) ws[i] = 0u;
}

// ---------------------------------------------------------------------------
// K1: build_target. One thread per batch element; sequential over the 20
// object slots reproduces the scan's last-write-wins semantics exactly.
// conf is SoA: conf[(bb*5+k)*NCELL + cell].
// ---------------------------------------------------------------------------
__global__ void k_build(const float* __restrict__ pred,
                        const float* __restrict__ target,
                        float* __restrict__ conf,
                        unsigned* __restrict__ clsmask,
                        int BS, int NCELL) {
  int b = blockIdx.x * blockDim.x + threadIdx.x;
  if (b >= BS) return;
  const float fg = (float)SGRID;
#pragma unroll 1
  for (int o = 0; o < MAXOBJ; ++o) {
    const float* t = target + (size_t)(b * MAXOBJ + o) * 5;
    float x1 = t[0], y1 = t[1], x2 = t[2], y2 = t[3], cls = t[4];
    if (x1 + y1 + x2 + y2 + cls == 0.0f) continue;  // invalid slot -> dropped
    float cx = (x1 + x2) * 0.5f * fg;
    float cy = (y1 + y2) * 0.5f * fg;
    float w = x2 - x1;
    float h = y2 - y1;
    int gx = (int)floorf(cx);
    gx = min(max(gx, 0), SGRID - 1);
    int gy = (int)floorf(cy);
    gy = min(max(gy, 0), SGRID - 1);
    float offx = cx - (float)gx;
    float offy = cy - (float)gy;

    const float* pb = pred + ((size_t)b * SS + (size_t)gy * SGRID + gx) * 30;
    float tcx = offx / fg, tcy = offy / fg;
    float t_lt_x = tcx - 0.5f * w, t_lt_y = tcy - 0.5f * h;
    float t_rb_x = tcx + 0.5f * w, t_rb_y = tcy + 0.5f * h;
    float a2 = (t_rb_x - t_lt_x) * (t_rb_y - t_lt_y);

    float iou0 = 0.0f, iou1 = 0.0f;
#pragma unroll
    for (int bb = 0; bb < 2; ++bb) {
      float px = pb[bb * 5 + 1] / fg;
      float py = pb[bb * 5 + 2] / fg;
      float pw = pb[bb * 5 + 3];
      float ph = pb[bb * 5 + 4];
      float p_lt_x = px - 0.5f * pw, p_lt_y = py - 0.5f * ph;
      float p_rb_x = px + 0.5f * pw, p_rb_y = py + 0.5f * ph;
      float lt_x = fmaxf(p_lt_x, t_lt_x), lt_y = fmaxf(p_lt_y, t_lt_y);
      float rb_x = fminf(p_rb_x, t_rb_x), rb_y = fminf(p_rb_y, t_rb_y);
      float wi = fmaxf(rb_x - lt_x, 0.0f);
      float hi = fmaxf(rb_y - lt_y, 0.0f);
      float inter = wi * hi;
      float a1 = (p_rb_x - p_lt_x) * (p_rb_y - p_lt_y);
      float iou = inter / (a1 + a2 - inter);
      if (bb == 0) iou0 = iou; else iou1 = iou;
    }
    int mi = (iou1 > iou0) ? 1 : 0;  // jnp.argmax: first max wins on ties

    int cell = b * SS + gy * SGRID + gx;
    conf[(mi * 5 + 0) * (size_t)NCELL + cell] = 1.0f;
    conf[(mi * 5 + 1) * (size_t)NCELL + cell] = offx;
    conf[(mi * 5 + 2) * (size_t)NCELL + cell] = offy;
    conf[(mi * 5 + 3) * (size_t)NCELL + cell] = w;
    conf[(mi * 5 + 4) * (size_t)NCELL + cell] = h;
    clsmask[cell] |= (1u << (int)cls);
  }
}

// ---------------------------------------------------------------------------
// K2: per-cell loss statistics -> deterministic block partials.
// acc: [0]=n_noobj [1]=noobj_sq [2]=n_resp [3]=objconf_sq
//      [4]=xy_sq   [5]=wh_sq    [6]=n_obj  [7]=clc_sq
// ---------------------------------------------------------------------------
__global__ void k_loss(const float* __restrict__ pred,
                       const float* __restrict__ conf,
                       const unsigned* __restrict__ clsmask,
                       float* __restrict__ part, int NCELL) {
  const int c = blockIdx.x * 256 + (int)threadIdx.x;
  float acc[8] = {0.f, 0.f, 0.f, 0.f, 0.f, 0.f, 0.f, 0.f};

  if (c < NCELL) {
    const float* p = pred + (size_t)c * 30;
    float cf[10];
#pragma unroll
    for (int k = 0; k < 10; ++k) cf[k] = conf[(size_t)k * NCELL + c];

    float confsum = cf[0] + cf[5];
    bool obj = (confsum == 1.0f);
    bool noobj = (confsum == 0.0f);

    if (noobj) {
      acc[0] = 2.0f;
      acc[1] = p[0] * p[0] + p[5] * p[5];
    }
#pragma unroll
    for (int bb = 0; bb < 2; ++bb) {
      float s = cf[bb * 5] + cf[bb * 5 + 1] + cf[bb * 5 + 2] + cf[bb * 5 + 3] +
                cf[bb * 5 + 4];
      bool resp = obj && (s != 0.0f);
      if (resp) {
        acc[2] += 1.0f;
        float dc = p[bb * 5 + 0] - cf[bb * 5 + 0];
        acc[3] += dc * dc;
        float dx = p[bb * 5 + 1] - cf[bb * 5 + 1];
        float dy = p[bb * 5 + 2] - cf[bb * 5 + 2];
        acc[4] += dx * dx + dy * dy;
        float dw = sqrtf(p[bb * 5 + 3]) - sqrtf(cf[bb * 5 + 3]);
        float dh = sqrtf(p[bb * 5 + 4]) - sqrtf(cf[bb * 5 + 4]);
        acc[5] += dw * dw + dh * dh;
      }
    }
    if (obj) {
      acc[6] = 1.0f;
      unsigned m = clsmask[c];
      float s = 0.0f;
#pragma unroll
      for (int cc = 0; cc < NCLS; ++cc) {
        float tgt = (float)((m >> cc) & 1u);
        float d = p[10 + cc] - tgt;
        s += d * d;
      }
      acc[7] = s;
    }
  }

  // Wave-level reduction via v_wmma_f32_16x16x4_f32 (all lanes active here).
  wave_reduce8_wmma(acc);

  __shared__ float lds[8 * 8];
  const int lane = (int)(threadIdx.x & 31u);
  const int wv = (int)(threadIdx.x >> 5u);
  if (lane == 0) {
#pragma unroll
    for (int a = 0; a < 8; ++a) lds[wv * 8 + a] = acc[a];
  }
  __syncthreads();
  if (threadIdx.x < 8) {
    float s = 0.0f;
#pragma unroll
    for (int w = 0; w < 8; ++w) s += lds[w * 8 + (int)threadIdx.x];
    part[(size_t)blockIdx.x * 8 + threadIdx.x] = s;
  }
}

// ---------------------------------------------------------------------------
// K3: single-wave final fold of block partials + loss formula.
// ---------------------------------------------------------------------------
__global__ void k_final(const float* __restrict__ part, int nblocks,
                        float* __restrict__ out) {
  float acc[8] = {0.f, 0.f, 0.f, 0.f, 0.f, 0.f, 0.f, 0.f};
  const int lane = (int)threadIdx.x;  // launched with 32 threads = 1 wave
  for (int i = lane; i < nblocks; i += 32) {
#pragma unroll
    for (int a = 0; a < 8; ++a) acc[a] += part[(size_t)i * 8 + a];
  }
  wave_reduce8_wmma(acc);
  if (lane == 0) {
    float n_noobj = acc[0], noobj_sq = acc[1];
    float n_resp = acc[2], oc_sq = acc[3];
    float xy_sq = acc[4], wh_sq = acc[5];
    float n_obj = acc[6], clc_sq = acc[7];
    float noobj_loss = noobj_sq / n_noobj;
    float objconf_loss = oc_sq / n_resp;
    float loss_xy = xy_sq / (n_resp * 2.0f);
    float loss_wh = wh_sq / (n_resp * 2.0f);
    float loss_clc = clc_sq / (n_obj * (float)NCLS);
    out[0] = 5.0f * (loss_xy + loss_wh) + objconf_loss + 0.5f * noobj_loss +
             loss_clc;
  }
}

// ---------------------------------------------------------------------------
extern "C" void kernel_launch(void* const* d_in, const int* in_sizes, int n_in,
                              void* d_out, int out_size, void* d_ws,
                              size_t ws_size, hipStream_t stream) {
  const float* pred = (const float*)d_in[0];    // (BS, 14, 14, 30) f32
  const float* target = (const float*)d_in[1];  // (BS, 20, 5) f32
  float* out = (float*)d_out;

  const int BS = in_sizes[1] / (MAXOBJ * 5);
  const int NCELL = BS * SS;
  const int nblocks = (NCELL + 255) / 256;

  // Workspace layout (all 4-byte aligned):
  float* conf = (float*)d_ws;                       // 10 * NCELL f32
  unsigned* clsmask = (unsigned*)(conf + (size_t)10 * NCELL);  // NCELL u32
  float* part = (float*)(clsmask + NCELL);          // nblocks * 8 f32

  const int zwords = 11 * NCELL;
  k_zero<<<(zwords + 255) / 256, 256, 0, stream>>>((unsigned*)d_ws, zwords);
  k_build<<<(BS + 255) / 256, 256, 0, stream>>>(pred, target, conf, clsmask,
                                                BS, NCELL);
  k_loss<<<nblocks, 256, 0, stream>>>(pred, conf, clsmask, part, NCELL);
  k_final<<<1, 32, 0, stream>>>(part, nblocks, out);
}